// NaiveLegendreKANLayer_712964571428
// MI455X (gfx1250) — compile-verified
//
#include <hip/hip_runtime.h>

#define INPUTDIM 512
#define OUTDIM   64
#define DEGREE   8
#define KDIM     (INPUTDIM * DEGREE)   // 4096
#define M_TILE   128                   // rows per workgroup (8 waves x 16 rows)
#define I_CHUNK  8                     // inputs staged per K-chunk
#define K_CHUNK  (I_CHUNK * DEGREE)    // 64 k-values per chunk
#define T_STEPS  (K_CHUNK / 4)         // 16 WMMA k-steps (K=4 each)
#define NCHUNKS  (INPUTDIM / I_CHUNK)  // 64

typedef __attribute__((ext_vector_type(2))) float v2f;
typedef __attribute__((ext_vector_type(8))) float v8f;

// ---- monotone float <-> uint mapping for atomic min/max ----
__device__ __forceinline__ unsigned f2ord(float f) {
    unsigned b = __float_as_uint(f);
    return (b & 0x80000000u) ? ~b : (b | 0x80000000u);
}
__device__ __forceinline__ float ord2f(unsigned u) {
    unsigned b = (u & 0x80000000u) ? (u & 0x7FFFFFFFu) : ~u;
    return __uint_as_float(b);
}

__global__ void kan_mm_init(unsigned* ws) {
    ws[0] = 0xFFFFFFFFu;  // min identity (ordered space)
    ws[1] = 0x00000000u;  // max identity
}

__global__ __launch_bounds__(256) void kan_minmax(const float4* __restrict__ xv,
                                                  unsigned* __restrict__ ws, int nvec) {
    float lmin = 3.402823466e38f, lmax = -3.402823466e38f;
    int stride = gridDim.x * blockDim.x;
    for (int i = blockIdx.x * blockDim.x + threadIdx.x; i < nvec; i += stride) {
        float4 v = xv[i];
        lmin = fminf(lmin, fminf(fminf(v.x, v.y), fminf(v.z, v.w)));
        lmax = fmaxf(lmax, fmaxf(fmaxf(v.x, v.y), fmaxf(v.z, v.w)));
    }
    __shared__ float smn[256], smx[256];
    int tid = threadIdx.x;
    smn[tid] = lmin; smx[tid] = lmax;
    __syncthreads();
    for (int s = 128; s > 0; s >>= 1) {
        if (tid < s) {
            smn[tid] = fminf(smn[tid], smn[tid + s]);
            smx[tid] = fmaxf(smx[tid], smx[tid + s]);
        }
        __syncthreads();
    }
    if (tid == 0) {
        atomicMin(&ws[0], f2ord(smn[0]));
        atomicMax(&ws[1], f2ord(smx[0]));
    }
}

// Main fused kernel: Legendre basis generation + f32 WMMA GEMM [65536 x 4096] x [4096 x 64]
__global__ __launch_bounds__(256) void kan_legendre_wmma(
    const float* __restrict__ x, const float* __restrict__ coeffs,
    const float* __restrict__ bias, const unsigned* __restrict__ mmws,
    float* __restrict__ out) {

    // A fragments pre-swizzled: [rowblock(8)][t(16)][lane(32)][2 floats]
    __shared__ __align__(16) float ldsA[8 * T_STEPS * 32 * 2];   // 32 KB
    // B fragments pre-swizzled: [t(16)][oblock(4)][lane(32)][2 floats]
    __shared__ __align__(16) float ldsB[T_STEPS * 4 * 32 * 2];   // 16 KB

    const int tid  = threadIdx.x;
    const int lane = tid & 31;
    const int wv   = tid >> 5;               // wave id = row-block 0..7
    const int wgM  = blockIdx.x * M_TILE;

    const float xmin = ord2f(mmws[0]);
    const float xmax = ord2f(mmws[1]);
    const float inv2 = 2.0f / (xmax - xmin);

    v8f acc0 = {0.f,0.f,0.f,0.f,0.f,0.f,0.f,0.f};
    v8f acc[4] = {acc0, acc0, acc0, acc0};

    // A-staging assignment: thread handles 4 (row,input) pairs = one float4 of x
    const int pm   = (tid * 4) >> 3;          // row within tile, 0..127
    const int pii0 = (tid * 4) & 7;           // local input base: 0 or 4
    // B-staging assignment: thread loads 16 consecutive k for one output column
    const int bo  = tid >> 2;                 // output column 0..63
    const int bk0 = (tid & 3) * 16;           // local k base

    for (int c = 0; c < NCHUNKS; ++c) {
        const int i0 = c * I_CHUNK;
        __syncthreads();  // previous chunk fully consumed

        // ---- stage A: compute Legendre basis, scatter into fragment layout ----
        float4 xv = *(const float4*)(x + (size_t)(wgM + pm) * INPUTDIM + i0 + pii0);
        float xs[4] = {xv.x, xv.y, xv.z, xv.w};
        #pragma unroll
        for (int j = 0; j < 4; ++j) {
            float xn = (xs[j] - xmin) * inv2 - 1.0f;
            float p0 = 1.0f;
            float p1 = xn;
            float p2 = ( 3.0f*xn*p1 - 1.0f*p0) * 0.5f;
            float p3 = ( 5.0f*xn*p2 - 2.0f*p1) * (1.0f/3.0f);
            float p4 = ( 7.0f*xn*p3 - 3.0f*p2) * 0.25f;
            float p5 = ( 9.0f*xn*p4 - 4.0f*p3) * 0.2f;
            float p6 = (11.0f*xn*p5 - 5.0f*p4) * (1.0f/6.0f);
            float p7 = (13.0f*xn*p6 - 6.0f*p5) * (1.0f/7.0f);
            float P[8] = {p0,p1,p2,p3,p4,p5,p6,p7};
            const int il = pii0 + j;
            #pragma unroll
            for (int d = 0; d < 8; ++d) {
                int kl   = il * 8 + d;                    // local k 0..63
                int t    = kl >> 2;
                int kq   = kl & 3;
                int dl   = (pm & 15) + ((kq & 2) << 3);   // lane: +16 for K2/K3
                int slot = kq & 1;                        // VGPR0 (K0/K2) or VGPR1 (K1/K3)
                ldsA[((((pm >> 4) * T_STEPS + t) * 32 + dl) << 1) + slot] = P[d];
            }
        }

        // ---- stage B: coeffs[o, i, d] -> B[k, o] fragments ----
        const float* cb = coeffs + (size_t)bo * KDIM + i0 * DEGREE + bk0;
        #pragma unroll
        for (int q = 0; q < 4; ++q) {
            float4 cv = *(const float4*)(cb + q * 4);
            float cs[4] = {cv.x, cv.y, cv.z, cv.w};
            #pragma unroll
            for (int j = 0; j < 4; ++j) {
                int kl   = bk0 + q * 4 + j;
                int t    = kl >> 2;
                int kq   = kl & 3;
                int dl   = (bo & 15) + ((kq & 2) << 3);
                int slot = kq & 1;
                ldsB[(((t * 4 + (bo >> 4)) * 32 + dl) << 1) + slot] = cs[j];
            }
        }
        __syncthreads();

        // ---- consume: 16 k-steps x 4 output blocks of v_wmma_f32_16x16x4_f32 ----
        #pragma unroll
        for (int t = 0; t < T_STEPS; ++t) {
            v2f a = *(const v2f*)&ldsA[((wv * T_STEPS + t) * 32 + lane) << 1];
            #pragma unroll
            for (int ob = 0; ob < 4; ++ob) {
                v2f b = *(const v2f*)&ldsB[((t * 4 + ob) * 32 + lane) << 1];
                acc[ob] = __builtin_amdgcn_wmma_f32_16x16x4_f32(
                    false, a, false, b, (short)0, acc[ob], false, false);
            }
        }
    }

    // ---- epilogue: bias + store (C layout: VGPR r -> rows r / r+8; lane -> col) ----
    const int n16   = lane & 15;
    const int mhalf = (lane >> 4) << 3;
    #pragma unroll
    for (int ob = 0; ob < 4; ++ob) {
        float bv = bias[ob * 16 + n16];
        #pragma unroll
        for (int r = 0; r < 8; ++r) {
            int row = wgM + wv * 16 + r + mhalf;
            out[(size_t)row * OUTDIM + ob * 16 + n16] = acc[ob][r] + bv;
        }
    }
}

extern "C" void kernel_launch(void* const* d_in, const int* in_sizes, int n_in,
                              void* d_out, int out_size, void* d_ws, size_t ws_size,
                              hipStream_t stream) {
    const float* x      = (const float*)d_in[0];
    const float* coeffs = (const float*)d_in[1];
    const float* bias   = (const float*)d_in[2];
    float* out          = (float*)d_out;
    unsigned* ws        = (unsigned*)d_ws;

    const int n = in_sizes[0];            // 16*4096*512
    const int nrows = n / INPUTDIM;       // 65536
    const int nvec = n / 4;

    kan_mm_init<<<1, 1, 0, stream>>>(ws);
    kan_minmax<<<2048, 256, 0, stream>>>((const float4*)x, ws, nvec);
    kan_legendre_wmma<<<nrows / M_TILE, 256, 0, stream>>>(x, coeffs, bias, ws, out);
}